// PatchMask_6004364279952
// MI455X (gfx1250) — compile-verified
//
#include <hip/hip_runtime.h>
#include <hip/hip_bf16.h>
#include <cstdint>

// Problem constants (fixed by the reference)
#define B_DIM 128
#define C_DIM 128
#define T_DIM 5000
#define PATCH_LEN 25
#define NUM_PATCHES 200                     // T / PATCH_LEN
#define TOTAL_UNITS (C_DIM * NUM_PATCHES)   // 25600
#define T4 (T_DIM / 4)                      // 1250 float4 per (b,c) row

typedef float v4f __attribute__((ext_vector_type(4)));

// ---------------------------------------------------------------------------
// Kernel 1: init unit mask (25600 floats in d_ws) to 1.0f
// ---------------------------------------------------------------------------
__global__ void pm_mask_init(float* __restrict__ umask) {
    int i = blockIdx.x * blockDim.x + threadIdx.x;
    if (i < TOTAL_UNITS) umask[i] = 1.0f;
}

// ---------------------------------------------------------------------------
// Kernel 2: scatter zeros at masked unit indices (idx entries are unique)
// ---------------------------------------------------------------------------
__global__ void pm_mask_scatter(float* __restrict__ umask,
                                const int* __restrict__ idx, int n) {
    int i = blockIdx.x * blockDim.x + threadIdx.x;
    if (i < n) umask[idx[i]] = 0.0f;
}

// ---------------------------------------------------------------------------
// Kernel 3: apply mask.
// grid = (5 t-tiles, 128 channels, 8 b-groups), block = 256 (8 wave32s).
// Stage this channel's 200-float mask row into LDS with the CDNA5 async
// global->LDS pipe, hoist 4 mask floats into registers, then stream 16 batch
// rows with NT b128 load / packed mul / NT b128 store.
// ---------------------------------------------------------------------------
__global__ void pm_apply(const float* __restrict__ x,
                         const float* __restrict__ umask,
                         float* __restrict__ out) {
    __shared__ __align__(16) float s_pmask[NUM_PATCHES]; // 800 bytes

    const int tid  = threadIdx.x;
    const int tile = blockIdx.x;        // 0..4   (t tiles of 256 float4)
    const int c    = blockIdx.y;        // 0..127 (channel)
    const int bg   = blockIdx.z;        // 0..7   (group of 16 batch rows)

    // ---- async DMA: 50 lanes pull 16B each -> 800B mask row into LDS ----
    if (tid < NUM_PATCHES / 4) {
        unsigned lds_off = (unsigned)(uintptr_t)(&s_pmask[tid * 4]);
        unsigned goff    = (unsigned)(tid * 16);
        unsigned long long src =
            (unsigned long long)(uintptr_t)(umask + (size_t)c * NUM_PATCHES);
        asm volatile(
            "global_load_async_to_lds_b128 %0, %1, %2"
            :
            : "v"(lds_off), "v"(goff), "s"(src)
            : "memory");
    }
    asm volatile("s_wait_asynccnt 0" ::: "memory");
    __syncthreads();

    const int f4 = tile * 256 + tid;    // float4 column within the row
    if (f4 >= T4) return;               // last tile is partial (226 of 256)

    const int t0 = f4 * 4;
    // Hoist the 4 per-element mask values into registers (reused for all 16 b).
    // Multiply (not select) so -0.0 for masked negative inputs matches the
    // reference bit-for-bit.
    v4f m;
    m.x = s_pmask[(t0 + 0) / PATCH_LEN];
    m.y = s_pmask[(t0 + 1) / PATCH_LEN];
    m.z = s_pmask[(t0 + 2) / PATCH_LEN];
    m.w = s_pmask[(t0 + 3) / PATCH_LEN];

    const v4f* __restrict__ xin = (const v4f*)x;
    v4f* __restrict__       o   = (v4f*)out;

    // index (in float4 units) of element (b0, c, t0)
    size_t pos = ((size_t)(bg * 16) * C_DIM + (size_t)c) * (size_t)T4 + (size_t)f4;
    const size_t bstride = (size_t)C_DIM * (size_t)T4;   // one batch row

    // warm the first line before entering the streaming loop
    __builtin_prefetch(&xin[pos], 0, 1);

#pragma unroll 8
    for (int i = 0; i < 16; ++i) {
        // x / out are touched exactly once: stream with non-temporal policy so
        // the 655 MB stream doesn't thrash L2 (keeps the mask table resident).
        v4f v = __builtin_nontemporal_load(&xin[pos]);
        v = v * m;
        __builtin_nontemporal_store(v, &o[pos]);
        pos += bstride;
    }
}

// ---------------------------------------------------------------------------
// Host-side launcher
// ---------------------------------------------------------------------------
extern "C" void kernel_launch(void* const* d_in, const int* in_sizes, int n_in,
                              void* d_out, int out_size, void* d_ws, size_t ws_size,
                              hipStream_t stream) {
    const float* x   = (const float*)d_in[0];
    const int*   idx = (const int*)d_in[1];
    const int num_mask = in_sizes[1];          // 2560
    float* out   = (float*)d_out;
    float* umask = (float*)d_ws;               // 25600 floats = 100 KB scratch

    // 1) unit mask = 1.0 everywhere
    pm_mask_init<<<(TOTAL_UNITS + 255) / 256, 256, 0, stream>>>(umask);
    // 2) zeros at masked units
    pm_mask_scatter<<<(num_mask + 255) / 256, 256, 0, stream>>>(umask, idx, num_mask);
    // 3) streaming masked multiply
    dim3 grid(5, C_DIM, 8);   // t-tiles x channels x batch-groups
    pm_apply<<<grid, 256, 0, stream>>>(x, umask, out);
}